// MoNet_33303176413655
// MI455X (gfx1250) — compile-verified
//
#include <hip/hip_runtime.h>
#include <math.h>

// MoNet / GMMConv, 2 layers. N=50000, E=800000, F=128, H=64, C=16, K=8, D=2.
// GEMMs on fp32 WMMA (V_WMMA_F32_16X16X4_F32) with fully compile-time strides
// (immediate-offset loads, no u64 address math). Gather/scatter phase is
// L2-resident (gx1 = 102 MB < 192 MB L2) and dominates; fp32 everywhere.

typedef float v2f __attribute__((ext_vector_type(2)));
typedef float v8f __attribute__((ext_vector_type(8)));

#define GMM_EPS 1e-15f

// ---------------------------------------------------------------------------
// GEMM: Out[M,NN] = A[M,KK] * W[KK,NN], row-major fp32, WMMA 16x16x4.
// Each wave computes a 16 x (16*NT) strip (NT accumulators share one A frag).
// M must be a multiple of 16 (50000 = 3125*16). All strides compile-time.
// ---------------------------------------------------------------------------
template<int KK, int NN, int NT>
__global__ __launch_bounds__(256)
void wmma_gemm_f32(const float* __restrict__ A, const float* __restrict__ W,
                   float* __restrict__ Out, int M) {
    constexpr int tilesN = NN / (16 * NT);
    const int lane   = threadIdx.x & 31;
    const int wave   = threadIdx.x >> 5;
    const int tileId = blockIdx.x * 8 + wave;
    const int tilesTotal = (M >> 4) * tilesN;
    if (tileId >= tilesTotal) return;

    const int tm = tileId / tilesN;
    const int tn = tileId - tm * tilesN;
    const int m0 = tm << 4;
    const int n0 = tn * (16 * NT);

    const int half = lane >> 4;      // 0: lanes 0-15, 1: lanes 16-31
    const int l15  = lane & 15;

    // A 16x4 frag: lane l<16 supplies row m0+l, K = kk,kk+1; lanes 16-31: kk+2,kk+3.
    const float* __restrict__ Ap = A + (size_t)(m0 + l15) * KK + half * 2;
    // B 4x16 frag: lane l<16 supplies col n+l rows kk,kk+1; lanes 16-31: kk+2,kk+3.
    const float* __restrict__ Wp = W + (size_t)(half * 2) * NN + n0 + l15;

    v8f acc[NT];
#pragma unroll
    for (int t = 0; t < NT; ++t) acc[t] = (v8f){};

#pragma unroll
    for (int kk = 0; kk < KK; kk += 4) {
        v2f a;
        a.x = Ap[kk];
        a.y = Ap[kk + 1];
#pragma unroll
        for (int t = 0; t < NT; ++t) {
            v2f b;
            b.x = Wp[kk * NN + t * 16];        // compile-time immediates
            b.y = Wp[kk * NN + NN + t * 16];
            acc[t] = __builtin_amdgcn_wmma_f32_16x16x4_f32(
                false, a, false, b, (short)0, acc[t], false, false);
        }
    }

    // C/D layout: VGPR j -> row m0+j (lanes 0-15) / m0+8+j (lanes 16-31).
    float* __restrict__ Op = Out + (size_t)(m0 + half * 8) * NN + n0 + l15;
#pragma unroll
    for (int t = 0; t < NT; ++t)
#pragma unroll
        for (int j = 0; j < 8; ++j)
            Op[j * NN + t * 16] = acc[t][j];
}

// ---------------------------------------------------------------------------
// Per-edge gaussian mixture weights: gauss[e,k] = exp(-0.5*sum_d diff^2/(eps+s^2))
// ---------------------------------------------------------------------------
__global__ __launch_bounds__(256)
void gauss_kernel(const float* __restrict__ attr, const float* __restrict__ mu,
                  const float* __restrict__ sigma, float* __restrict__ gauss, int E) {
    int e = blockIdx.x * blockDim.x + threadIdx.x;
    if (e >= E) return;
    float p0 = attr[2 * e], p1 = attr[2 * e + 1];
#pragma unroll
    for (int k = 0; k < 8; ++k) {
        float d0 = p0 - mu[2 * k];
        float d1 = p1 - mu[2 * k + 1];
        float s0 = sigma[2 * k], s1 = sigma[2 * k + 1];
        float q  = d0 * d0 / (GMM_EPS + s0 * s0) + d1 * d1 / (GMM_EPS + s1 * s1);
        gauss[8 * e + k] = expf(-0.5f * q);
    }
}

// ---------------------------------------------------------------------------
// Layer-1 message + scatter: one wave per edge, H=64 (2 floats/lane).
// msg = sum_k gauss[e,k]*gx[row[e],k,:]; atomic scatter into agg[col[e],:].
// Also accumulates degree count (shared by both layers).
// ---------------------------------------------------------------------------
__global__ __launch_bounds__(256)
void edge_scatter_h64(const long long* __restrict__ ei, const float* __restrict__ gauss,
                      const float* __restrict__ gx, float* __restrict__ agg,
                      float* __restrict__ cnt, int E) {
    const int lane = threadIdx.x & 31;
    const int e    = (int)((blockIdx.x * blockDim.x + threadIdx.x) >> 5);
    if (e >= E) return;
    const long long row = ei[e];
    const long long col = ei[(size_t)E + e];
    const float* __restrict__ g  = gauss + (size_t)e * 8;
    const float* __restrict__ gr = gx + (size_t)row * 512 + lane;   // K*H = 8*64
    float a0 = 0.f, a1 = 0.f;
#pragma unroll
    for (int k = 0; k < 8; ++k) {
        float w = g[k];
        a0 = fmaf(w, gr[k * 64],      a0);
        a1 = fmaf(w, gr[k * 64 + 32], a1);
    }
    atomicAdd(&agg[(size_t)col * 64 + lane],      a0);
    atomicAdd(&agg[(size_t)col * 64 + 32 + lane], a1);
    if (lane == 0) atomicAdd(&cnt[col], 1.0f);
}

// ---------------------------------------------------------------------------
// Layer-2 message + scatter: 2 edges per wave, C=16 (1 float per 16-lane half).
// ---------------------------------------------------------------------------
__global__ __launch_bounds__(256)
void edge_scatter_h16(const long long* __restrict__ ei, const float* __restrict__ gauss,
                      const float* __restrict__ gx, float* __restrict__ agg, int E) {
    const int lane = threadIdx.x & 31;
    const int sub  = lane & 15;
    const int e    = (int)(((blockIdx.x * blockDim.x + threadIdx.x) >> 5) * 2 + (lane >> 4));
    if (e >= E) return;
    const long long row = ei[e];
    const long long col = ei[(size_t)E + e];
    const float* __restrict__ g  = gauss + (size_t)e * 8;
    const float* __restrict__ gr = gx + (size_t)row * 128 + sub;    // K*C = 8*16
    float a = 0.f;
#pragma unroll
    for (int k = 0; k < 8; ++k) a = fmaf(g[k], gr[k * 16], a);
    atomicAdd(&agg[(size_t)col * 16 + sub], a);
}

// ---------------------------------------------------------------------------
// Finalize layer 1: h = elu(agg/max(cnt,1) + root + bias), H=64.
// ---------------------------------------------------------------------------
__global__ __launch_bounds__(256)
void finalize1(const float* __restrict__ agg, const float* __restrict__ cnt,
               const float* __restrict__ r1, const float* __restrict__ bias,
               float* __restrict__ h, int N) {
    int t = blockIdx.x * blockDim.x + threadIdx.x;
    if (t >= N * 64) return;
    int n = t >> 6, j = t & 63;
    float v = agg[t] / fmaxf(cnt[n], 1.0f) + r1[t] + bias[j];
    h[t] = v > 0.f ? v : expm1f(v);
}

// ---------------------------------------------------------------------------
// Finalize layer 2 + log_softmax over C=16 classes: 2 nodes per wave,
// 16-lane shuffle reductions for max and sum-exp.
// ---------------------------------------------------------------------------
__global__ __launch_bounds__(256)
void finalize2(const float* __restrict__ agg, const float* __restrict__ cnt,
               const float* __restrict__ r2, const float* __restrict__ bias,
               float* __restrict__ out, int N) {
    const int lane = threadIdx.x & 31;
    const int sub  = lane & 15;
    const int n    = (int)(((blockIdx.x * blockDim.x + threadIdx.x) >> 5) * 2 + (lane >> 4));
    if (n >= N) return;
    float v = agg[(size_t)n * 16 + sub] / fmaxf(cnt[n], 1.0f)
            + r2[(size_t)n * 16 + sub] + bias[sub];
    float m = v;
#pragma unroll
    for (int off = 8; off >= 1; off >>= 1) m = fmaxf(m, __shfl_xor(m, off, 16));
    float s = expf(v - m);
#pragma unroll
    for (int off = 8; off >= 1; off >>= 1) s += __shfl_xor(s, off, 16);
    out[(size_t)n * 16 + sub] = v - m - logf(s);
}

__global__ __launch_bounds__(256)
void fill_zero(float* __restrict__ p, size_t n) {
    size_t t = (size_t)blockIdx.x * blockDim.x + threadIdx.x;
    if (t < n) p[t] = 0.f;
}

// ---------------------------------------------------------------------------
extern "C" void kernel_launch(void* const* d_in, const int* in_sizes, int n_in,
                              void* d_out, int out_size, void* d_ws, size_t ws_size,
                              hipStream_t stream) {
    const float*     x    = (const float*)d_in[0];
    const long long* ei   = (const long long*)d_in[1];   // int64 [2,E]
    const float*     attr = (const float*)d_in[2];
    const float*     g1w  = (const float*)d_in[3];
    const float*     mu1  = (const float*)d_in[4];
    const float*     sg1  = (const float*)d_in[5];
    const float*     r1w  = (const float*)d_in[6];
    const float*     b1   = (const float*)d_in[7];
    const float*     g2w  = (const float*)d_in[8];
    const float*     mu2  = (const float*)d_in[9];
    const float*     sg2  = (const float*)d_in[10];
    const float*     r2w  = (const float*)d_in[11];
    const float*     b2   = (const float*)d_in[12];
    float* out = (float*)d_out;

    const int F = 128;                       // Fin
    const int N = in_sizes[0] / F;           // 50000
    const int E = in_sizes[1] / 2;           // 800000

    // Workspace carve-up (floats). ~225 MB total.
    float* ws   = (float*)d_ws;
    float* gx1  = ws;  ws += (size_t)N * 512;   // x @ g1_w       [N, K*H]
    float* r1   = ws;  ws += (size_t)N * 64;    // x @ root1_w    [N, H]
    float* agg1 = ws;  ws += (size_t)N * 64;
    float* cnt  = ws;  ws += (size_t)N;
    float* h    = ws;  ws += (size_t)N * 64;
    float* gx2  = ws;  ws += (size_t)N * 128;   // h @ g2_w       [N, K*C]
    float* r2   = ws;  ws += (size_t)N * 16;    // h @ root2_w    [N, C]
    float* agg2 = ws;  ws += (size_t)N * 16;
    float* gs1  = ws;  ws += (size_t)E * 8;
    float* gs2  = ws;  ws += (size_t)E * 8;

    // Zero accumulators.
    {
        size_t n1 = (size_t)N * 64, n2 = (size_t)N, n3 = (size_t)N * 16;
        fill_zero<<<(unsigned)((n1 + 255) / 256), 256, 0, stream>>>(agg1, n1);
        fill_zero<<<(unsigned)((n2 + 255) / 256), 256, 0, stream>>>(cnt,  n2);
        fill_zero<<<(unsigned)((n3 + 255) / 256), 256, 0, stream>>>(agg2, n3);
    }

    // ---- Layer 1 ----
    {   // x[N,128] @ g1_w[128,512] -> gx1 ; 2 tiles/wave
        int tiles = (N / 16) * (512 / 32);
        wmma_gemm_f32<128, 512, 2><<<(tiles + 7) / 8, 256, 0, stream>>>(x, g1w, gx1, N);
    }
    {   // x[N,128] @ root1_w[128,64] -> r1
        int tiles = (N / 16) * (64 / 32);
        wmma_gemm_f32<128, 64, 2><<<(tiles + 7) / 8, 256, 0, stream>>>(x, r1w, r1, N);
    }
    gauss_kernel<<<(E + 255) / 256, 256, 0, stream>>>(attr, mu1, sg1, gs1, E);
    edge_scatter_h64<<<(E + 7) / 8, 256, 0, stream>>>(ei, gs1, gx1, agg1, cnt, E);
    finalize1<<<(N * 64 + 255) / 256, 256, 0, stream>>>(agg1, cnt, r1, b1, h, N);

    // ---- Layer 2 ----
    {   // h[N,64] @ g2_w[64,128] -> gx2
        int tiles = (N / 16) * (128 / 32);
        wmma_gemm_f32<64, 128, 2><<<(tiles + 7) / 8, 256, 0, stream>>>(h, g2w, gx2, N);
    }
    {   // h[N,64] @ root2_w[64,16] -> r2 ; NN=16 -> single tile per wave
        int tiles = (N / 16) * (16 / 16);
        wmma_gemm_f32<64, 16, 1><<<(tiles + 7) / 8, 256, 0, stream>>>(h, r2w, r2, N);
    }
    gauss_kernel<<<(E + 255) / 256, 256, 0, stream>>>(attr, mu2, sg2, gs2, E);
    {
        int waves = (E + 1) / 2;
        edge_scatter_h16<<<(waves + 7) / 8, 256, 0, stream>>>(ei, gs2, gx2, agg2, E);
    }
    finalize2<<<(N + 15) / 16, 256, 0, stream>>>(agg2, cnt, r2, b2, out, N);
}